// HeterGraphAttentionLayer_80633716015374
// MI455X (gfx1250) — compile-verified
//
#include <hip/hip_runtime.h>
#include <math.h>

// ---- problem constants (from reference) ----
#define NA   35          // N_AQI
#define NM   18          // N_MEO
#define NN   53          // N = NA + NM
#define FO   64          // F_OUT
#define CTXF 60          // CTX
#define DF   124         // F_OUT + CTX

typedef float v2f __attribute__((ext_vector_type(2)));
typedef float v8f __attribute__((ext_vector_type(8)));

// LDS layout (floats).  hf/attri have 68 rows so unconditional 16-row WMMA tile
// stores at rowBase=35 (max row 66) stay in bounds.
#define OFF_HF     0                       // 68 x 128 (cols 0..63 heter feat, 64..123 ctx)
#define OFF_ATTRI  (OFF_HF    + 68*128)    // 68 x 64  heter_attri (rows 53.. zeroed by meo GEMM pads)
#define OFF_ATTN   (OFF_ATTRI + 68*64)     // 64 x 64  attention / output staging
#define OFF_AFULL  (OFF_ATTN  + 64*64)     // 48 x 32  aqi_full (zero padded)
#define OFF_MFULL  (OFF_AFULL + 48*32)     // 32 x 32  meo_full (zero padded)
#define OFF_WUA    (OFF_MFULL + 32*32)     // 32 x 64  W_ua (rows 24..31 zero)
#define OFF_WUM    (OFF_WUA   + 32*64)     // 32 x 64  W_um (rows 26..31 zero)
#define OFF_WXA    (OFF_WUM   + 32*64)     // 16 x 64
#define OFF_WXM    (OFF_WXA   + 16*64)     // 16 x 64
#define OFF_AVEC   (OFF_WXM   + 16*64)     // 4 x 256  a_aa,a_am,a_ma,a_mm (249 used)
#define OFF_S      (OFF_AVEC  + 4*256)     // 64 x 2
#define OFF_T      (OFF_S     + 128)       // 64 x 2
#define SMEM_FLOATS (OFF_T + 128)

// One 16x16 f32 tile, K = 4*ksteps. A(MxK) row-major lda, B(KxN) row-major ldb.
// v_wmma_f32_16x16x4_f32 operand layout: A 16x4 -> 2 VGPR (lanes0-15: K=0,1;
// lanes16-31: K=2,3); B 4x16 mirrored; C/D standard 8-VGPR layout.
__device__ __forceinline__ v8f wmma_mm(const float* __restrict__ A, int lda,
                                       const float* __restrict__ Bm, int ldb,
                                       int m0, int n0, int ksteps, int lane)
{
    v8f acc = {};
    const int mr = lane & 15;
    const int kh = (lane >> 4) << 1;           // 0 or 2
    const float* ap = A  + (m0 + mr) * lda + kh;       // 8B aligned (lda even, kh even)
    const float* bp = Bm + kh * ldb + (n0 + mr);
    for (int ks = 0; ks < ksteps; ++ks) {
        v2f av = *(const v2f*)ap;              // ds_load_b64
        v2f bv;
        bv.x = bp[0];  bv.y = bp[ldb];         // pairable ds_load_2addr
        acc = __builtin_amdgcn_wmma_f32_16x16x4_f32(false, av, false, bv,
                                                    (short)0, acc, false, false);
        ap += 4;
        bp += 4 * ldb;
    }
    return acc;
}

// Unconditional 16x16 D-tile store (dest rows rowBase+m0 .. rowBase+m0+15).
__device__ __forceinline__ void store_tile(float* __restrict__ D, int ldd, v8f acc,
                                           int m0, int n0, int rowBase, int lane)
{
    float* p = D + (rowBase + m0 + ((lane >> 4) << 3)) * ldd + (n0 + (lane & 15));
#pragma unroll
    for (int v = 0; v < 8; ++v) { *p = acc[v]; p += ldd; }
}

extern "C" __global__ __launch_bounds__(256)
void hga_fused_kernel(const float* __restrict__ aqi_inp, const float* __restrict__ meo_inp,
                      const float* __restrict__ ctx,     const float* __restrict__ adjn,
                      const float* __restrict__ idE,     const float* __restrict__ monE,
                      const float* __restrict__ wdE,     const float* __restrict__ hrE,
                      const float* __restrict__ mwE,     const float* __restrict__ midE,
                      const float* __restrict__ mmonE,   const float* __restrict__ mwdE,
                      const float* __restrict__ mhrE,
                      const float* __restrict__ Wxa,     const float* __restrict__ Wxm,
                      const float* __restrict__ Wua,     const float* __restrict__ Wum,
                      const float* __restrict__ a_aa,    const float* __restrict__ a_am,
                      const float* __restrict__ a_ma,    const float* __restrict__ a_mm,
                      const int*   __restrict__ aqi_ex,  const int* __restrict__ meo_ex,
                      const int*   __restrict__ adj,
                      float* __restrict__ out, int batch)
{
    extern __shared__ float sm[];
    float* hf    = sm + OFF_HF;
    float* attri = sm + OFF_ATTRI;
    float* attn  = sm + OFF_ATTN;
    float* afull = sm + OFF_AFULL;
    float* mfull = sm + OFF_MFULL;
    float* wua   = sm + OFF_WUA;
    float* wum   = sm + OFF_WUM;
    float* wxa   = sm + OFF_WXA;
    float* wxm   = sm + OFF_WXM;
    float* avec  = sm + OFF_AVEC;
    float* sS    = sm + OFF_S;
    float* tT    = sm + OFF_T;

    const int tid  = threadIdx.x;
    const int lane = tid & 31;
    const int wave = tid >> 5;
    const int b    = blockIdx.x;

    const float* ain  = aqi_inp + (size_t)b * NA * 16;
    const float* minp = meo_inp + (size_t)b * NM * 16;
    const int*   aex  = aqi_ex  + (size_t)b * NA * 4;
    const int*   mex  = meo_ex  + (size_t)b * NM * 5;

    __builtin_prefetch(ain,  0, 0);   // global_prefetch_b8
    __builtin_prefetch(minp, 0, 0);

    // ---------------- Phase A: stage weights / context / inputs into LDS ----------------

    // context_feat -> strided hf slots via gfx1250 async global->LDS DMA (GVS mode:
    // per-lane LDS dest VGPR + per-lane 32-bit src offset + SGPR base).
    {
        const unsigned ldsBase = (unsigned)((uintptr_t)hf - (uintptr_t)sm)
                               + __builtin_amdgcn_groupstaticsize();
        for (int e = tid; e < NN * CTXF; e += 256) {
            const int r = e / CTXF, c = e - r * CTXF;
            const unsigned dst = ldsBase + (unsigned)((r * 128 + 64 + c) * 4);
            const unsigned src = (unsigned)e * 4u;
            asm volatile("global_load_async_to_lds_b32 %0, %1, %2 offset:0"
                         :: "v"(dst), "v"(src), "s"(ctx) : "memory");
        }
    }

    // weights (vectorized float4 copies)
    {
        const float4* Wxa4 = (const float4*)Wxa;
        const float4* Wxm4 = (const float4*)Wxm;
        float4* wxa4 = (float4*)wxa;
        float4* wxm4 = (float4*)wxm;
        for (int k = tid; k < 16 * 16; k += 256) { wxa4[k] = Wxa4[k]; wxm4[k] = Wxm4[k]; }

        const float4* Wua4 = (const float4*)Wua;
        const float4* Wum4 = (const float4*)Wum;
        float4* wua4 = (float4*)wua;
        float4* wum4 = (float4*)wum;
        const float4 z4 = make_float4(0.f, 0.f, 0.f, 0.f);
        for (int k = tid; k < 32 * 16; k += 256) {       // r = k>>4 (16 float4 per row)
            const int r = k >> 4;
            wua4[k] = (r < 24) ? Wua4[k] : z4;
            wum4[k] = (r < 26) ? Wum4[k] : z4;
        }
    }
    for (int k = tid; k < 4 * 256; k += 256) {
        const int v = k >> 8, i = k & 255;
        const float* av = (v == 0) ? a_aa : (v == 1) ? a_am : (v == 2) ? a_ma : a_mm;
        avec[k] = (i < 249) ? av[i] : 0.f;
    }
    // zero the pad slots of the hf context half (rows >=53 or cols >=60)
    for (int k = tid; k < 64 * 64; k += 256) {
        const int r = k >> 6, c = k & 63;
        if (r >= NN || c >= CTXF) hf[r * 128 + 64 + c] = 0.f;
    }
    for (int k = tid; k < 48 * 32; k += 256) {       // aqi_full = [inp(16) | 4 embeds(2)]
        const int m = k >> 5, c = k & 31;
        float v = 0.f;
        if (m < NA) {
            if (c < 16) v = ain[m * 16 + c];
            else if (c < 24) {
                const int p = (c - 16) >> 1, comp = (c - 16) & 1;
                const int idx = aex[m * 4 + p];
                const float* tab = (p == 0) ? idE : (p == 1) ? monE : (p == 2) ? wdE : hrE;
                v = tab[idx * 2 + comp];
            }
        }
        afull[k] = v;
    }
    for (int k = tid; k < 32 * 32; k += 256) {       // meo_full = [inp(16) | 5 embeds(2)]
        const int m = k >> 5, c = k & 31;
        float v = 0.f;
        if (m < NM) {
            if (c < 16) v = minp[m * 16 + c];
            else if (c < 26) {
                const int p = (c - 16) >> 1, comp = (c - 16) & 1;
                const int idx = mex[m * 5 + p];
                const float* tab = (p == 0) ? mwE : (p == 1) ? midE :
                                   (p == 2) ? mmonE : (p == 3) ? mwdE : mhrE;
                v = tab[idx * 2 + comp];
            }
        }
        mfull[k] = v;
    }
    asm volatile("s_wait_asynccnt 0x0" ::: "memory");   // async ctx DMA complete
    __syncthreads();

    // ---------------- Phase B1: aqi projection GEMMs (store rows 0..47 unconditionally) ----
    // attri[0..47] = aqi_inp @ W_xa     (A = afull cols 0..15, zero-padded rows)
    for (int t = wave; t < 12; t += 8) {
        const int mt = t >> 2, nt = t & 3;
        v8f acc = wmma_mm(afull, 32, wxa, 64, mt * 16, nt * 16, 4, lane);
        store_tile(attri, 64, acc, mt * 16, nt * 16, 0, lane);
    }
    // hf[0..47, 0:64] = aqi_full @ W_ua (K padded to 32)
    for (int t = wave; t < 12; t += 8) {
        const int mt = t >> 2, nt = t & 3;
        v8f acc = wmma_mm(afull, 32, wua, 64, mt * 16, nt * 16, 8, lane);
        store_tile(hf, 128, acc, mt * 16, nt * 16, 0, lane);
    }
    __syncthreads();

    // ---------------- Phase B2: meo projection GEMMs overwrite rows 35..66 ----------------
    // (zero A-pad rows write the zeros rows 53..63 need for the final GEMM's K-pad)
    for (int t = wave; t < 8; t += 8) {
        const int mt = t >> 2, nt = t & 3;
        v8f acc = wmma_mm(mfull, 32, wxm, 64, mt * 16, nt * 16, 4, lane);
        store_tile(attri, 64, acc, mt * 16, nt * 16, NA, lane);
    }
    for (int t = wave; t < 8; t += 8) {
        const int mt = t >> 2, nt = t & 3;
        v8f acc = wmma_mm(mfull, 32, wum, 64, mt * 16, nt * 16, 8, lane);
        store_tile(hf, 128, acc, mt * 16, nt * 16, NA, lane);
    }
    __syncthreads();

    // ---------------- Phase C: attention logit dot products s,t (float4 vectorized) ------
    // sS[i][dstType], tT[j][srcType]; avec order: 0=aa 1=am 2=ma 3=mm
    for (int task = tid; task < NN * 4; task += 256) {
        const int row = task >> 2;
        const int d   = task & 1;
        const int isT = (task >> 1) & 1;
        int vec;
        if (!isT) vec = (row < NA) ? (d ? 1 : 0) : (d ? 3 : 2);
        else      vec = (row < NA) ? (d ? 2 : 0) : (d ? 3 : 1);
        const float4* hr4 = (const float4*)(hf + row * 128);
        const float4* av4 = (const float4*)(avec + vec * 256 + (isT ? DF : 0)); // 496B: 16B-aligned
        float acc = 0.f;
        for (int i = 0; i < DF / 4; ++i) {
            const float4 h = hr4[i], a = av4[i];
            acc = fmaf(h.x, a.x, fmaf(h.y, a.y, fmaf(h.z, a.z, fmaf(h.w, a.w, acc))));
        }
        if (!isT) sS[row * 2 + d] = acc;
        else      tT[row * 2 + d] = acc;
    }
    __syncthreads();

    // ---------------- Phase D: e + leaky_relu + mask + softmax (row per wave) -------------
    const float c_aa = avec[0 * 256 + 248], c_am = avec[1 * 256 + 248];
    const float c_ma = avec[2 * 256 + 248], c_mm = avec[3 * 256 + 248];
    for (int i = wave; i < NN; i += 8) {
        const int srcT = (i >= NA);
        const int j0 = lane, j1 = lane + 32;
        float x0, x1 = -1e30f;
        {
            const int dstT = (j0 >= NA);
            const float cs = srcT ? (dstT ? c_mm : c_ma) : (dstT ? c_am : c_aa);
            float e = sS[i * 2 + dstT] + tT[j0 * 2 + srcT] + adjn[i * NN + j0] * cs;
            e = (e > 0.f) ? e : 0.2f * e;
            x0 = (adj[i * NN + j0] > 0) ? e : -1e12f;
        }
        if (j1 < NN) {
            const int dstT = (j1 >= NA);
            const float cs = srcT ? (dstT ? c_mm : c_ma) : (dstT ? c_am : c_aa);
            float e = sS[i * 2 + dstT] + tT[j1 * 2 + srcT] + adjn[i * NN + j1] * cs;
            e = (e > 0.f) ? e : 0.2f * e;
            x1 = (adj[i * NN + j1] > 0) ? e : -1e12f;
        }
        float mx = fmaxf(x0, x1);
        for (int m = 16; m >= 1; m >>= 1) mx = fmaxf(mx, __shfl_xor(mx, m, 32));
        const float p0 = __expf(x0 - mx);
        const float p1 = (j1 < NN) ? __expf(x1 - mx) : 0.f;
        float ssum = p0 + p1;
        for (int m = 16; m >= 1; m >>= 1) ssum += __shfl_xor(ssum, m, 32);
        const float inv = 1.f / ssum;
        attn[i * 64 + j0] = p0 * inv;
        attn[i * 64 + j1] = p1 * inv;     // zero pad cols 53..63
    }
    __syncthreads();

    // ---------------- Phase E: heter_out = attention @ heter_attri ------------------------
    // attn M-pad rows 53..63 are uninitialized: they only affect D rows 53..63,
    // which are never stored; K-pads are zero (attn cols + attri rows 53..63).
    const int tA = wave, tB = wave + 8;   // 16 output tiles, 2 per wave
    v8f accA = wmma_mm(attn, 64, attri, 64, (tA >> 2) * 16, (tA & 3) * 16, 16, lane);
    v8f accB = wmma_mm(attn, 64, attri, 64, (tB >> 2) * 16, (tB & 3) * 16, 16, lane);
    __syncthreads();                       // all reads of attn done -> reuse as staging
    store_tile(attn, 64, accA, (tA >> 2) * 16, (tA & 3) * 16, 0, lane);
    store_tile(attn, 64, accB, (tB >> 2) * 16, (tB & 3) * 16, 0, lane);
    __syncthreads();

    // coalesced float4 stores, split into the two output tensors
    float* out0 = out;
    float* out1 = out + (size_t)batch * NA * FO;
    for (int k = tid; k < NN * 16; k += 256) {
        const int r = k >> 4, c4 = (k & 15) << 2;
        const float4 v = *(const float4*)&attn[r * 64 + c4];
        float* dst = (r < NA) ? (out0 + ((size_t)b * NA + r) * FO + c4)
                              : (out1 + ((size_t)b * NM + (r - NA)) * FO + c4);
        *(float4*)dst = v;
    }
}

extern "C" void kernel_launch(void* const* d_in, const int* in_sizes, int n_in,
                              void* d_out, int out_size, void* d_ws, size_t ws_size,
                              hipStream_t stream)
{
    const float* aqi_inp = (const float*)d_in[0];
    const float* meo_inp = (const float*)d_in[1];
    const float* ctx     = (const float*)d_in[2];
    const float* adjn    = (const float*)d_in[3];
    const float* idE     = (const float*)d_in[4];
    const float* monE    = (const float*)d_in[5];
    const float* wdE     = (const float*)d_in[6];
    const float* hrE     = (const float*)d_in[7];
    const float* mwE     = (const float*)d_in[8];
    const float* midE    = (const float*)d_in[9];
    const float* mmonE   = (const float*)d_in[10];
    const float* mwdE    = (const float*)d_in[11];
    const float* mhrE    = (const float*)d_in[12];
    const float* Wxa     = (const float*)d_in[13];
    const float* Wxm     = (const float*)d_in[14];
    const float* Wua     = (const float*)d_in[15];
    const float* Wum     = (const float*)d_in[16];
    const float* a_aa    = (const float*)d_in[17];
    const float* a_am    = (const float*)d_in[18];
    const float* a_ma    = (const float*)d_in[19];
    const float* a_mm    = (const float*)d_in[20];
    const int*   aqi_ex  = (const int*)d_in[21];
    const int*   meo_ex  = (const int*)d_in[22];
    const int*   adj     = (const int*)d_in[23];

    const int batch = in_sizes[0] / (NA * 16);
    const size_t shmem = (size_t)SMEM_FLOATS * sizeof(float);

    hga_fused_kernel<<<batch, 256, shmem, stream>>>(
        aqi_inp, meo_inp, ctx, adjn,
        idE, monE, wdE, hrE,
        mwE, midE, mmonE, mwdE, mhrE,
        Wxa, Wxm, Wua, Wum,
        a_aa, a_am, a_ma, a_mm,
        aqi_ex, meo_ex, adj,
        (float*)d_out, batch);
}